// GATModel_28793460752452
// MI455X (gfx1250) — compile-verified
//
#include <hip/hip_runtime.h>
#include <math.h>

// ---------------------------------------------------------------------------
// GAT (2 layers, heads=1) + L2 norm + TopKPooling for gfx1250.
// Edge aggregation is bandwidth/atomic bound (~3 GB traffic over 3.3M edges);
// the dense GEMMs use V_WMMA_F32_16X16X4_F32 (fp32-exact CDNA5 matrix op).
// ---------------------------------------------------------------------------

typedef __attribute__((ext_vector_type(2))) float v2f;
typedef __attribute__((ext_vector_type(8))) float v8f;

#define NEG_SLOPE 0.2f
#define EPSN 1e-12f

__device__ __forceinline__ unsigned mono_key(float f) {
  unsigned u = __float_as_uint(f);
  return u ^ ((u >> 31) ? 0xFFFFFFFFu : 0x80000000u);
}
__device__ __forceinline__ float inv_mono(unsigned k) {
  unsigned b = (k & 0x80000000u) ? (k ^ 0x80000000u) : ~k;
  return __uint_as_float(b);
}
__device__ __forceinline__ float lrelu(float x) { return x > 0.f ? x : NEG_SLOPE * x; }

// ------------------------- GEMM 1: [N,128] @ [128,16] -----------------------
// One wave computes one 16x16 output tile via 32 k-steps of WMMA f32 16x16x4.
__global__ void gemm_x_w1(const float* __restrict__ X, const float* __restrict__ W,
                          float* __restrict__ H, int Nn) {
  int wave = blockIdx.x * (blockDim.x >> 5) + (threadIdx.x >> 5);
  int lane = threadIdx.x & 31;
  int half = lane >> 4, l15 = lane & 15;
  int rowTiles = (Nn + 15) >> 4;
  if (wave >= rowTiles) return;
  int row0 = wave * 16;
  v8f c = {};
  int mA = row0 + l15; if (mA >= Nn) mA = Nn - 1;
  const float* xrow = X + (size_t)mA * 128;
  for (int kb = 0; kb < 128; kb += 4) {
    v2f a, b;
    // A 16x4: lanes0-15 -> K = kb+0,kb+1 ; lanes16-31 -> K = kb+2,kb+3
    a.x = xrow[kb + 2 * half];
    a.y = xrow[kb + 2 * half + 1];
    // B 4x16: row-major W[k][n], n = l15
    b.x = W[(kb + 2 * half) * 16 + l15];
    b.y = W[(kb + 2 * half + 1) * 16 + l15];
    c = __builtin_amdgcn_wmma_f32_16x16x4_f32(false, a, false, b, (short)0, c, false, false);
  }
  for (int r = 0; r < 8; ++r) {
    int m = row0 + r + 8 * half;   // C/D: vgpr r, halves hold M=r and M=r+8
    if (m < Nn) H[(size_t)m * 16 + l15] = c[r];
  }
}

// ------------------------- GEMM 2: [N,16] @ [16,64] -------------------------
__global__ void gemm_h_w3(const float* __restrict__ Hn, const float* __restrict__ W,
                          float* __restrict__ H2, int Nn) {
  int rowTiles = (Nn + 15) >> 4;
  int wave = blockIdx.x * (blockDim.x >> 5) + (threadIdx.x >> 5);
  int lane = threadIdx.x & 31;
  int half = lane >> 4, l15 = lane & 15;
  int rt = wave >> 2, ct = wave & 3;
  if (rt >= rowTiles) return;
  int row0 = rt * 16, n0 = ct * 16;
  v8f c = {};
  int mA = row0 + l15; if (mA >= Nn) mA = Nn - 1;
  const float* hrow = Hn + (size_t)mA * 16;
  for (int kb = 0; kb < 16; kb += 4) {
    v2f a, b;
    a.x = hrow[kb + 2 * half];
    a.y = hrow[kb + 2 * half + 1];
    b.x = W[(kb + 2 * half) * 64 + n0 + l15];
    b.y = W[(kb + 2 * half + 1) * 64 + n0 + l15];
    c = __builtin_amdgcn_wmma_f32_16x16x4_f32(false, a, false, b, (short)0, c, false, false);
  }
  for (int r = 0; r < 8; ++r) {
    int m = row0 + r + 8 * half;
    if (m < Nn) H2[(size_t)m * 64 + n0 + l15] = c[r];
  }
}

// ------------------------- per-node attention scores ------------------------
template <int F>
__global__ void node_scores(const float* __restrict__ H, const float* __restrict__ as,
                            const float* __restrict__ ad, float* __restrict__ s,
                            float* __restrict__ t, int Nn) {
  int i = blockIdx.x * blockDim.x + threadIdx.x;
  if (i >= Nn) return;
  const float* h = H + (size_t)i * F;
  float ss = 0.f, tt = 0.f;
  for (int f = 0; f < F; ++f) { float v = h[f]; ss += v * as[f]; tt += v * ad[f]; }
  s[i] = ss;
  t[i] = tt;
}

// ------------------------- edge passes (segmented softmax) ------------------
__global__ void edge_max(const int* __restrict__ src, const int* __restrict__ dst,
                         const float* __restrict__ s, const float* __restrict__ t,
                         unsigned* __restrict__ mkey, long long E, long long Etot) {
  long long e = (long long)blockIdx.x * blockDim.x + threadIdx.x;
  if (e >= Etot) return;
  int u, v;
  if (e < E) { u = src[e]; v = dst[e]; } else { u = v = (int)(e - E); }
  float l = lrelu(s[u] + t[v]);
  atomicMax(&mkey[v], mono_key(l));   // monotonic uint transform; init 0 < any finite
}

__global__ void edge_expsum(const int* __restrict__ src, const int* __restrict__ dst,
                            const float* __restrict__ s, const float* __restrict__ t,
                            const unsigned* __restrict__ mkey, float* __restrict__ sum,
                            float* __restrict__ ev, long long E, long long Etot) {
  long long e = (long long)blockIdx.x * blockDim.x + threadIdx.x;
  if (e >= Etot) return;
  int u, v;
  if (e < E) { u = src[e]; v = dst[e]; } else { u = v = (int)(e - E); }
  float l = lrelu(s[u] + t[v]);
  float m = inv_mono(mkey[v]);
  float x = expf(l - m);
  ev[e] = x;
  atomicAdd(&sum[v], x);
}

template <int F>
__global__ void edge_scatter(const int* __restrict__ src, const int* __restrict__ dst,
                             const float* __restrict__ H, const float* __restrict__ ev,
                             const float* __restrict__ sum, float* __restrict__ acc,
                             long long E, long long Etot) {
  const int CH = F / 4;                                 // float4 chunks per edge
  long long t_ = (long long)blockIdx.x * blockDim.x + threadIdx.x;
  long long total = Etot * CH;
  if (t_ >= total) return;
  long long e = t_ / CH;
  int c = (int)(t_ - e * CH);
  int u, v;
  if (e < E) { u = src[e]; v = dst[e]; } else { u = v = (int)(e - E); }
  float w = ev[e] / sum[v];
  const float4* hv = (const float4*)(H + (size_t)u * F);
  float4 hx = hv[c];
  float* ap = acc + (size_t)v * F + c * 4;
  atomicAdd(ap + 0, hx.x * w);
  atomicAdd(ap + 1, hx.y * w);
  atomicAdd(ap + 2, hx.z * w);
  atomicAdd(ap + 3, hx.w * w);
}

// ------------------------- bias (+relu) + L2 normalize ----------------------
template <int F, bool RELU>
__global__ void finalize_norm(float* __restrict__ A, const float* __restrict__ b, int Nn) {
  int i = blockIdx.x * blockDim.x + threadIdx.x;
  if (i >= Nn) return;
  float* a = A + (size_t)i * F;
  float reg[F];
  float nrm = 0.f;
  for (int f = 0; f < F; ++f) {
    float v = a[f] + b[f];
    if (RELU) v = fmaxf(v, 0.f);
    reg[f] = v;
    nrm += v * v;
  }
  float inv = 1.f / fmaxf(sqrtf(nrm), EPSN);
  for (int f = 0; f < F; ++f) a[f] = reg[f] * inv;
}

// ------------------------- pooling score ------------------------------------
__global__ void score_kernel(const float* __restrict__ H2, const float* __restrict__ pw,
                             float* __restrict__ score, int Nn) {
  int i = blockIdx.x * blockDim.x + threadIdx.x;
  if (i >= Nn) return;
  float pn = 0.f;
  for (int f = 0; f < 64; ++f) pn += pw[f] * pw[f];
  pn = fmaxf(sqrtf(pn), EPSN);
  const float* h = H2 + (size_t)i * 64;
  float d = 0.f;
  for (int f = 0; f < 64; ++f) d += h[f] * pw[f];
  score[i] = tanhf(d / pn);
}

// ------------------------- stable LSD radix sort (descending) ---------------
__global__ void sort_init(const float* __restrict__ score, unsigned* __restrict__ key,
                          unsigned* __restrict__ val, int Nn) {
  int i = blockIdx.x * blockDim.x + threadIdx.x;
  if (i >= Nn) return;
  key[i] = ~mono_key(score[i]);     // ascending sort of ~mono == descending score
  val[i] = (unsigned)i;
}

__global__ void radix_hist(const unsigned* __restrict__ key, unsigned* __restrict__ hist,
                           int n, int shift, int numBlocks) {
  __shared__ unsigned h[256];
  int tid = threadIdx.x;            // blockDim == 256
  h[tid] = 0;
  __syncthreads();
  int i = blockIdx.x * 256 + tid;
  if (i < n) atomicAdd(&h[(key[i] >> shift) & 255], 1u);
  __syncthreads();
  hist[tid * numBlocks + blockIdx.x] = h[tid];   // bin-major for stable global scan
}

__global__ void radix_scan(unsigned* __restrict__ hist, int B) {
  __shared__ unsigned sums[1024];
  int tid = threadIdx.x;            // blockDim == 1024, single block
  int chunk = (B + 1023) / 1024;
  int lo = tid * chunk;
  int hi = lo + chunk; if (hi > B) hi = B;
  unsigned s = 0;
  for (int i = lo; i < hi; ++i) s += hist[i];
  sums[tid] = s;
  __syncthreads();
  if (tid == 0) {
    unsigned run = 0;
    for (int i = 0; i < 1024; ++i) { unsigned tv = sums[i]; sums[i] = run; run += tv; }
  }
  __syncthreads();
  unsigned run = sums[tid];
  for (int i = lo; i < hi; ++i) { unsigned tv = hist[i]; hist[i] = run; run += tv; }
}

__global__ void radix_scatter(const unsigned* __restrict__ keyIn, const unsigned* __restrict__ valIn,
                              unsigned* __restrict__ keyOut, unsigned* __restrict__ valOut,
                              const unsigned* __restrict__ hist, int n, int shift, int numBlocks) {
  __shared__ unsigned char bins[256];
  int tid = threadIdx.x;            // blockDim == 256
  int i = blockIdx.x * 256 + tid;
  int cnt = n - blockIdx.x * 256; if (cnt > 256) cnt = 256;
  unsigned k = 0, pv = 0, b = 0;
  if (tid < cnt) {
    k = keyIn[i]; pv = valIn[i];
    b = (k >> shift) & 255;
    bins[tid] = (unsigned char)b;
  }
  __syncthreads();
  if (tid < cnt) {
    int rank = 0;                   // stable intra-block rank
    for (int j = 0; j < tid; ++j)
      if (bins[j] == (unsigned char)b) ++rank;
    unsigned pos = hist[b * numBlocks + blockIdx.x] + rank;
    keyOut[pos] = k;
    valOut[pos] = pv;
  }
}

// ------------------------- gather top-k output ------------------------------
__global__ void topk_gather(const unsigned* __restrict__ idxSorted, const float* __restrict__ score,
                            const float* __restrict__ H2, float* __restrict__ out, int k) {
  long long t = (long long)blockIdx.x * blockDim.x + threadIdx.x;
  if (t >= (long long)k * 64) return;
  int i = (int)(t >> 6), f = (int)(t & 63);
  unsigned idx = idxSorted[i];
  out[t] = H2[(size_t)idx * 64 + f] * score[idx];
}

// ---------------------------------------------------------------------------
extern "C" void kernel_launch(void* const* d_in, const int* in_sizes, int n_in,
                              void* d_out, int out_size, void* d_ws, size_t ws_size,
                              hipStream_t stream) {
  const float* x      = (const float*)d_in[0];
  const int*   ei     = (const int*)d_in[1];
  const float* W1     = (const float*)d_in[2];
  const float* a_src1 = (const float*)d_in[3];
  const float* a_dst1 = (const float*)d_in[4];
  const float* b1     = (const float*)d_in[5];
  const float* W3     = (const float*)d_in[6];
  const float* a_src3 = (const float*)d_in[7];
  const float* a_dst3 = (const float*)d_in[8];
  const float* b3     = (const float*)d_in[9];
  const float* pool_w = (const float*)d_in[10];

  const int Nn = in_sizes[0] / 128;
  const long long E = in_sizes[1] / 2;
  const long long Etot = E + Nn;
  const int* srcI = ei;
  const int* dstI = ei + E;

  // ---- carve workspace ----
  char* p = (char*)d_ws;
  auto alloc = [&](size_t bytes) -> void* {
    void* r = (void*)p;
    p += (bytes + 255) & ~(size_t)255;
    return r;
  };
  float*    h1    = (float*)alloc((size_t)Nn * 16 * sizeof(float));
  float*    acc1  = (float*)alloc((size_t)Nn * 16 * sizeof(float));  // -> h1n
  float*    h2    = (float*)alloc((size_t)Nn * 64 * sizeof(float));
  float*    acc2  = (float*)alloc((size_t)Nn * 64 * sizeof(float));  // -> h2n
  float*    s1    = (float*)alloc((size_t)Nn * sizeof(float));
  float*    t1    = (float*)alloc((size_t)Nn * sizeof(float));
  unsigned* m1key = (unsigned*)alloc((size_t)Nn * sizeof(unsigned));
  float*    sum1  = (float*)alloc((size_t)Nn * sizeof(float));
  float*    s2    = (float*)alloc((size_t)Nn * sizeof(float));
  float*    t2    = (float*)alloc((size_t)Nn * sizeof(float));
  unsigned* m2key = (unsigned*)alloc((size_t)Nn * sizeof(unsigned));
  float*    sum2  = (float*)alloc((size_t)Nn * sizeof(float));
  float*    ev    = (float*)alloc((size_t)Etot * sizeof(float));
  float*    score = (float*)alloc((size_t)Nn * sizeof(float));
  unsigned* keyA  = (unsigned*)alloc((size_t)Nn * sizeof(unsigned));
  unsigned* valA  = (unsigned*)alloc((size_t)Nn * sizeof(unsigned));
  unsigned* keyB  = (unsigned*)alloc((size_t)Nn * sizeof(unsigned));
  unsigned* valB  = (unsigned*)alloc((size_t)Nn * sizeof(unsigned));
  const int sortBlocks = (Nn + 255) / 256;
  unsigned* hist = (unsigned*)alloc((size_t)256 * sortBlocks * sizeof(unsigned));
  (void)ws_size; (void)n_in;

  auto cdiv = [](long long a, long long b) -> unsigned { return (unsigned)((a + b - 1) / b); };
  const unsigned rowTiles = (unsigned)((Nn + 15) / 16);

  // ---- layer 1 ----
  gemm_x_w1<<<cdiv(rowTiles, 8), 256, 0, stream>>>(x, W1, h1, Nn);
  node_scores<16><<<cdiv(Nn, 256), 256, 0, stream>>>(h1, a_src1, a_dst1, s1, t1, Nn);
  hipMemsetAsync(m1key, 0, (size_t)Nn * 4, stream);
  hipMemsetAsync(sum1, 0, (size_t)Nn * 4, stream);
  hipMemsetAsync(acc1, 0, (size_t)Nn * 16 * 4, stream);
  edge_max<<<cdiv(Etot, 256), 256, 0, stream>>>(srcI, dstI, s1, t1, m1key, E, Etot);
  edge_expsum<<<cdiv(Etot, 256), 256, 0, stream>>>(srcI, dstI, s1, t1, m1key, sum1, ev, E, Etot);
  edge_scatter<16><<<cdiv(Etot * 4, 256), 256, 0, stream>>>(srcI, dstI, h1, ev, sum1, acc1, E, Etot);
  finalize_norm<16, true><<<cdiv(Nn, 256), 256, 0, stream>>>(acc1, b1, Nn);   // relu + l2norm

  // ---- layer 2 ----
  gemm_h_w3<<<cdiv((long long)rowTiles * 4, 8), 256, 0, stream>>>(acc1, W3, h2, Nn);
  node_scores<64><<<cdiv(Nn, 256), 256, 0, stream>>>(h2, a_src3, a_dst3, s2, t2, Nn);
  hipMemsetAsync(m2key, 0, (size_t)Nn * 4, stream);
  hipMemsetAsync(sum2, 0, (size_t)Nn * 4, stream);
  hipMemsetAsync(acc2, 0, (size_t)Nn * 64 * 4, stream);
  edge_max<<<cdiv(Etot, 256), 256, 0, stream>>>(srcI, dstI, s2, t2, m2key, E, Etot);
  edge_expsum<<<cdiv(Etot, 256), 256, 0, stream>>>(srcI, dstI, s2, t2, m2key, sum2, ev, E, Etot);
  edge_scatter<64><<<cdiv(Etot * 16, 256), 256, 0, stream>>>(srcI, dstI, h2, ev, sum2, acc2, E, Etot);
  finalize_norm<64, false><<<cdiv(Nn, 256), 256, 0, stream>>>(acc2, b3, Nn);  // l2norm

  // ---- top-k pooling ----
  score_kernel<<<cdiv(Nn, 256), 256, 0, stream>>>(acc2, pool_w, score, Nn);
  sort_init<<<cdiv(Nn, 256), 256, 0, stream>>>(score, keyA, valA, Nn);
  const int Bins = 256 * sortBlocks;
  for (int pass = 0; pass < 4; ++pass) {
    unsigned* ki = (pass & 1) ? keyB : keyA;
    unsigned* vi = (pass & 1) ? valB : valA;
    unsigned* ko = (pass & 1) ? keyA : keyB;
    unsigned* vo = (pass & 1) ? valA : valB;
    int shift = 8 * pass;
    radix_hist<<<sortBlocks, 256, 0, stream>>>(ki, hist, Nn, shift, sortBlocks);
    radix_scan<<<1, 1024, 0, stream>>>(hist, Bins);
    radix_scatter<<<sortBlocks, 256, 0, stream>>>(ki, vi, ko, vo, hist, Nn, shift, sortBlocks);
  }
  // after 4 passes (A->B->A->B->A) result is in keyA/valA
  const int k = (Nn + 1) / 2;
  topk_gather<<<cdiv((long long)k * 64, 256), 256, 0, stream>>>(valA, score, acc2,
                                                                (float*)d_out, k);
}